// HeteroGNN_35390530519883
// MI455X (gfx1250) — compile-verified
//
#include <hip/hip_runtime.h>
#include <math.h>

typedef __attribute__((ext_vector_type(2))) float v2f;
typedef __attribute__((ext_vector_type(4))) float v4f;   // plain vector, AS-friendly
typedef __attribute__((ext_vector_type(8))) float v8f;

// Explicit global (address_space(1)) pointer types so VMEM ops lower to
// global_load_*/global_store_*/global_atomic_* instead of flat_* (flat ticks
// both LOADcnt and DScnt on CDNA5; global only LOADcnt/STOREcnt).
typedef const __attribute__((address_space(1))) float* gcf_t;
typedef       __attribute__((address_space(1))) float* gf_t;
typedef const __attribute__((address_space(1))) v4f*   gcv4_t;
typedef       __attribute__((address_space(1))) v4f*   gv4_t;
typedef const __attribute__((address_space(1))) int*   gci_t;

#define NP 50000
#define NG 50000
#define HDIM 128
#define NLAYERS 3

__device__ __forceinline__ void gatomic_add(float* p, float v) {
    __hip_atomic_fetch_add((gf_t)p, v, __ATOMIC_RELAXED, __HIP_MEMORY_SCOPE_AGENT);
}

// ---------------------------------------------------------------------------
// Edge scatter: one wave32 per edge; lane l handles features [4l, 4l+4).
// agg[dst] += x_src[src]; cnt[dst] += 1 (lane 0). Feature matrices (25.6MB)
// are L2-resident on MI455X (192MB L2), so these atomics stay on-chip.
// ---------------------------------------------------------------------------
__global__ void scatter_mean_accum(const float* __restrict__ xsrc,
                                   const int* __restrict__ esrc,
                                   const int* __restrict__ edst,
                                   float* __restrict__ agg,
                                   float* __restrict__ cnt,
                                   int E)
{
    int t = blockIdx.x * blockDim.x + threadIdx.x;
    int e = t >> 5;
    int lane = t & 31;
    if (e >= E) return;
    int s = ((gci_t)esrc)[e];
    int d = ((gci_t)edst)[e];
    v4f v = ((gcv4_t)(xsrc + (size_t)s * HDIM))[lane];
    float* dst = agg + (size_t)d * HDIM + lane * 4;
    gatomic_add(dst + 0, v.x);
    gatomic_add(dst + 1, v.y);
    gatomic_add(dst + 2, v.z);
    gatomic_add(dst + 3, v.w);
    if (lane == 0) gatomic_add(cnt + d, 1.0f);
}

// agg[row,:] /= max(cnt[row], 1)
__global__ void normalize_rows(float* __restrict__ agg,
                               const float* __restrict__ cnt,
                               int nrows)
{
    int idx = blockIdx.x * blockDim.x + threadIdx.x;   // one v4f each
    int total = nrows * (HDIM / 4);
    if (idx >= total) return;
    int row = idx / (HDIM / 4);
    float inv = 1.0f / fmaxf(((gcf_t)cnt)[row], 1.0f);
    gv4_t p = ((gv4_t)agg) + idx;
    v4f v = *p;
    v *= inv;
    *p = v;
}

// ---------------------------------------------------------------------------
// One A@W accumulation pass over K=128 with V_WMMA_F32_16X16X4_F32.
// A-lane layout (ISA 7.12.2, 32-bit A 16x4): lanes 0-15 carry M=lane,{K,K+1};
// lanes 16-31 carry M=lane-16,{K+2,K+3}. B (4x16) mirrors it. The wave feeds
// TWO adjacent 16-col output tiles from one A fetch (2 WMMAs per A load).
// ---------------------------------------------------------------------------
__device__ __forceinline__ void mma_pair(gcf_t A, gcf_t Wc0, gcf_t Wc1,
                                         int twoHalf, v8f& acc0, v8f& acc1)
{
#pragma unroll
    for (int k = 0; k < HDIM; k += 4) {
        v2f a, b0, b1;
        a.x  = A[k + twoHalf];
        a.y  = A[k + twoHalf + 1];
        b0.x = Wc0[(k + twoHalf) * HDIM];
        b0.y = Wc0[(k + twoHalf + 1) * HDIM];
        b1.x = Wc1[(k + twoHalf) * HDIM];
        b1.y = Wc1[(k + twoHalf + 1) * HDIM];
        acc0 = __builtin_amdgcn_wmma_f32_16x16x4_f32(
            false, a, false, b0, (short)0, acc0, false, false);
        acc1 = __builtin_amdgcn_wmma_f32_16x16x4_f32(
            false, a, false, b1, (short)0, acc1, false, false);
    }
}

// ---------------------------------------------------------------------------
// Fused SAGE GEMM: out = sum_{p<nPairs} A_p @ W_p + bias0 + bias1, optional
// leaky-ReLU. Block = 128 threads = 4 wave32s; blockIdx.x selects a 16-row
// stripe (50000 = 3125*16, no tail); wave w owns cols [32w, 32w+32) as two
// 16-wide tiles sharing one A fetch stream.
// ---------------------------------------------------------------------------
__global__ void fused_sage_gemm(const float* __restrict__ A0, const float* __restrict__ W0,
                                const float* __restrict__ A1, const float* __restrict__ W1,
                                const float* __restrict__ A2, const float* __restrict__ W2,
                                const float* __restrict__ A3, const float* __restrict__ W3,
                                const float* __restrict__ bias0,
                                const float* __restrict__ bias1,
                                float* __restrict__ out,
                                int nPairs, int applyLeaky)
{
    const int lane = threadIdx.x & 31;
    const int wave = threadIdx.x >> 5;   // 0..3
    const int m0 = blockIdx.x * 16;
    const int n0 = wave * 32;            // two col tiles: n0, n0+16
    const int half = lane >> 4;          // 0: lanes 0-15, 1: lanes 16-31
    const int l15 = lane & 15;
    const int twoHalf = 2 * half;

    const size_t rowOff = (size_t)(m0 + l15) * HDIM;
    const int    colOff = n0 + l15;

    v8f acc0 = {}, acc1 = {};

    mma_pair((gcf_t)(A0 + rowOff), (gcf_t)(W0 + colOff), (gcf_t)(W0 + colOff + 16),
             twoHalf, acc0, acc1);
    if (nPairs > 1)
        mma_pair((gcf_t)(A1 + rowOff), (gcf_t)(W1 + colOff), (gcf_t)(W1 + colOff + 16),
                 twoHalf, acc0, acc1);
    if (nPairs > 2) {
        mma_pair((gcf_t)(A2 + rowOff), (gcf_t)(W2 + colOff), (gcf_t)(W2 + colOff + 16),
                 twoHalf, acc0, acc1);
        mma_pair((gcf_t)(A3 + rowOff), (gcf_t)(W3 + colOff), (gcf_t)(W3 + colOff + 16),
                 twoHalf, acc0, acc1);
    }

    float bcol0 = 0.0f, bcol1 = 0.0f;
    if (bias0) { bcol0 += ((gcf_t)bias0)[colOff]; bcol1 += ((gcf_t)bias0)[colOff + 16]; }
    if (bias1) { bcol0 += ((gcf_t)bias1)[colOff]; bcol1 += ((gcf_t)bias1)[colOff + 16]; }

    gf_t o = (gf_t)out;
#pragma unroll
    for (int v = 0; v < 8; ++v) {
        const size_t row = (size_t)(m0 + v + 8 * half) * HDIM;
        float v0 = acc0[v] + bcol0;
        float v1 = acc1[v] + bcol1;
        if (applyLeaky) {
            v0 = (v0 > 0.0f) ? v0 : 0.01f * v0;
            v1 = (v1 > 0.0f) ? v1 : 0.01f * v1;
        }
        o[row + colOff]      = v0;
        o[row + colOff + 16] = v1;
    }
}

// ---------------------------------------------------------------------------
// Edge decoder: one wave32 per labeled edge; v4f partial dot per lane,
// 5-step shfl_xor tree reduction over 32 lanes (wave32!), sigmoid.
// ---------------------------------------------------------------------------
__global__ void edge_decoder(const float* __restrict__ xp,
                             const float* __restrict__ xg,
                             const int* __restrict__ ei,
                             const int* __restrict__ ej,
                             float* __restrict__ out,
                             int E)
{
    int t = blockIdx.x * blockDim.x + threadIdx.x;
    int e = t >> 5;
    int lane = t & 31;
    if (e >= E) return;
    int i = ((gci_t)ei)[e];
    int j = ((gci_t)ej)[e];
    v4f a = ((gcv4_t)(xp + (size_t)i * HDIM))[lane];
    v4f b = ((gcv4_t)(xg + (size_t)j * HDIM))[lane];
    float s = a.x * b.x + a.y * b.y + a.z * b.z + a.w * b.w;
#pragma unroll
    for (int off = 16; off > 0; off >>= 1)
        s += __shfl_xor(s, off, 32);
    if (lane == 0) ((gf_t)out)[e] = 1.0f / (1.0f + __expf(-s));
}

// ---------------------------------------------------------------------------
extern "C" void kernel_launch(void* const* d_in, const int* in_sizes, int n_in,
                              void* d_out, int out_size, void* d_ws, size_t ws_size,
                              hipStream_t stream)
{
    (void)n_in; (void)out_size; (void)ws_size;

    const float* x_pheno = (const float*)d_in[0];
    const float* x_gene  = (const float*)d_in[1];
    const float* Wl_isa  = (const float*)d_in[2];
    const float* bl_isa  = (const float*)d_in[3];
    const float* Wr_isa  = (const float*)d_in[4];
    const float* Wl_rel  = (const float*)d_in[5];
    const float* bl_rel  = (const float*)d_in[6];
    const float* Wr_rel  = (const float*)d_in[7];
    const float* Wl_rev  = (const float*)d_in[8];
    const float* bl_rev  = (const float*)d_in[9];
    const float* Wr_rev  = (const float*)d_in[10];
    const int* edge_isa  = (const int*)d_in[11];
    const int* edge_rel  = (const int*)d_in[12];
    const int* edge_rev  = (const int*)d_in[13];
    const int* eli       = (const int*)d_in[14];

    const int E_isa = in_sizes[11] / 2;
    const int E_rel = in_sizes[12] / 2;
    const int E_rev = in_sizes[13] / 2;
    const int E_lab = in_sizes[14] / 2;

    const size_t feat = (size_t)NP * HDIM;   // NP == NG
    float* ws = (float*)d_ws;
    float* xp_a    = ws; ws += feat;
    float* xp_b    = ws; ws += feat;
    float* xg_a    = ws; ws += feat;
    float* xg_b    = ws; ws += feat;
    float* agg_isa = ws; ws += feat;
    float* agg_rev = ws; ws += feat;
    float* agg_rel = ws; ws += feat;
    float* cnt_isa = ws; ws += NP;
    float* cnt_rev = ws; ws += NP;
    float* cnt_rel = ws; ws += NG;

    (void)hipMemcpyAsync(xp_a, x_pheno, feat * sizeof(float),
                         hipMemcpyDeviceToDevice, stream);
    (void)hipMemcpyAsync(xg_a, x_gene, feat * sizeof(float),
                         hipMemcpyDeviceToDevice, stream);

    float* xp_cur = xp_a; float* xp_new = xp_b;
    float* xg_cur = xg_a; float* xg_new = xg_b;

    const dim3 blk(256);
    const dim3 gblk(128);   // 4 wave32s per GEMM block

    for (int i = 0; i < NLAYERS; ++i) {
        // agg + cnt buffers are contiguous: clear all in one memset
        (void)hipMemsetAsync(agg_isa, 0,
                             (3 * feat + (size_t)(2 * NP + NG)) * sizeof(float), stream);

        scatter_mean_accum<<<(E_isa * 32 + 255) / 256, blk, 0, stream>>>(
            xp_cur, edge_isa, edge_isa + E_isa, agg_isa, cnt_isa, E_isa);
        scatter_mean_accum<<<(E_rev * 32 + 255) / 256, blk, 0, stream>>>(
            xg_cur, edge_rev, edge_rev + E_rev, agg_rev, cnt_rev, E_rev);
        scatter_mean_accum<<<(E_rel * 32 + 255) / 256, blk, 0, stream>>>(
            xp_cur, edge_rel, edge_rel + E_rel, agg_rel, cnt_rel, E_rel);

        const int nrm_blocks = (NP * (HDIM / 4) + 255) / 256;
        normalize_rows<<<nrm_blocks, blk, 0, stream>>>(agg_isa, cnt_isa, NP);
        normalize_rows<<<nrm_blocks, blk, 0, stream>>>(agg_rev, cnt_rev, NP);
        normalize_rows<<<nrm_blocks, blk, 0, stream>>>(agg_rel, cnt_rel, NG);

        const size_t wo = (size_t)i * HDIM * HDIM;
        const size_t bo = (size_t)i * HDIM;
        const int leaky = (i < NLAYERS - 1) ? 1 : 0;

        // new_p = agg_isa@Wl_isa + xp@Wr_isa + agg_rev@Wl_rev + xp@Wr_rev + bl_isa + bl_rev
        fused_sage_gemm<<<NP / 16, gblk, 0, stream>>>(
            agg_isa, Wl_isa + wo, xp_cur, Wr_isa + wo,
            agg_rev, Wl_rev + wo, xp_cur, Wr_rev + wo,
            bl_isa + bo, bl_rev + bo, xp_new, 4, leaky);

        // new_g = agg_rel@Wl_rel + xg@Wr_rel + bl_rel
        fused_sage_gemm<<<NG / 16, gblk, 0, stream>>>(
            agg_rel, Wl_rel + wo, xg_cur, Wr_rel + wo,
            nullptr, nullptr, nullptr, nullptr,
            bl_rel + bo, nullptr, xg_new, 2, leaky);

        float* t;
        t = xp_cur; xp_cur = xp_new; xp_new = t;
        t = xg_cur; xg_cur = xg_new; xg_new = t;
    }

    edge_decoder<<<(E_lab * 32 + 255) / 256, blk, 0, stream>>>(
        xp_cur, xg_cur, eli, eli + E_lab, (float*)d_out, E_lab);
}